// PartFlowNet_28097676051193
// MI455X (gfx1250) — compile-verified
//
#include <hip/hip_runtime.h>
#include <hip/hip_bf16.h>

typedef __attribute__((ext_vector_type(2))) float v2f;
typedef __attribute__((ext_vector_type(8))) float v8f;

#define B_SZ 8
#define N0   4096
#define FPS_T 256
#define TROWS 128   // rows per block in the GEMM kernel (4 waves x 2 row-tiles)

// ---------------------------------------------------------------------------
// f0 = concat(mask, normals)  -> (B,N,4)
// ---------------------------------------------------------------------------
__global__ void buildf0_kernel(const float* __restrict__ mask,
                               const float* __restrict__ nrm,
                               float* __restrict__ f0, int total) {
  int t = blockIdx.x * blockDim.x + threadIdx.x;
  if (t >= total) return;
  f0[t * 4 + 0] = mask[t];
  f0[t * 4 + 1] = nrm[t * 3 + 0];
  f0[t * 4 + 2] = nrm[t * 3 + 1];
  f0[t * 4 + 3] = nrm[t * 3 + 2];
}

// ---------------------------------------------------------------------------
// Farthest point sampling. One block per batch. Deterministic start at idx 0,
// argmax tie-break = lowest index (matches jnp.argmax).
// ---------------------------------------------------------------------------
__global__ __launch_bounds__(FPS_T) void fps_kernel(const float* __restrict__ pts,
                                                    float* __restrict__ cent,
                                                    int N, int S) {
  __shared__ float rv[FPS_T];
  __shared__ int   ri[FPS_T];
  __shared__ float nxt[3];
  const int b = blockIdx.x, t = threadIdx.x;
  const float* P = pts + (size_t)b * N * 3;
  float* C = cent + (size_t)b * S * 3;
  float x0 = P[0], y0 = P[1], z0 = P[2];
  if (t == 0) { C[0] = x0; C[1] = y0; C[2] = z0; }
  float mind[16];
#pragma unroll
  for (int i = 0; i < 16; ++i) {
    int j = t + i * FPS_T;
    if (j < N) {
      float dx = P[j * 3] - x0, dy = P[j * 3 + 1] - y0, dz = P[j * 3 + 2] - z0;
      mind[i] = dx * dx + dy * dy + dz * dz;
    } else mind[i] = -1.0f;
  }
  for (int s = 1; s < S; ++s) {
    float bv = -1.0f; int bj = 0x7fffffff;
#pragma unroll
    for (int i = 0; i < 16; ++i) {
      int j = t + i * FPS_T;
      if (j < N && mind[i] > bv) { bv = mind[i]; bj = j; }
    }
    rv[t] = bv; ri[t] = bj;
    __syncthreads();
    for (int off = FPS_T / 2; off > 0; off >>= 1) {
      if (t < off) {
        float ov = rv[t + off]; int oj = ri[t + off];
        if (ov > rv[t] || (ov == rv[t] && oj < ri[t])) { rv[t] = ov; ri[t] = oj; }
      }
      __syncthreads();
    }
    if (t == 0) {
      int j = ri[0];
      nxt[0] = P[j * 3]; nxt[1] = P[j * 3 + 1]; nxt[2] = P[j * 3 + 2];
      C[s * 3 + 0] = nxt[0]; C[s * 3 + 1] = nxt[1]; C[s * 3 + 2] = nxt[2];
    }
    __syncthreads();
    float nx = nxt[0], ny = nxt[1], nz = nxt[2];
#pragma unroll
    for (int i = 0; i < 16; ++i) {
      int j = t + i * FPS_T;
      if (j < N) {
        float dx = P[j * 3] - nx, dy = P[j * 3 + 1] - ny, dz = P[j * 3 + 2] - nz;
        mind[i] = fminf(mind[i], dx * dx + dy * dy + dz * dz);
      }
    }
    __syncthreads();
  }
}

// ---------------------------------------------------------------------------
// Ball query: one wave32 per (b,s). First-K in index order, -1 padded.
// Wave-level compaction via ballot + prefix popcount.
// ---------------------------------------------------------------------------
__global__ __launch_bounds__(256) void ballq_kernel(const float* __restrict__ cent,
                                                    const float* __restrict__ pts,
                                                    int* __restrict__ idx,
                                                    int total, int S, int N,
                                                    float r2, int K) {
  int wid = blockIdx.x * (blockDim.x >> 5) + (threadIdx.x >> 5);
  int lane = threadIdx.x & 31;
  if (wid >= total) return;          // uniform per wave
  int b = wid / S;
  const float* P = pts + (size_t)b * N * 3;
  float cx = cent[(size_t)wid * 3 + 0];
  float cy = cent[(size_t)wid * 3 + 1];
  float cz = cent[(size_t)wid * 3 + 2];
  int* out = idx + (size_t)wid * K;
  for (int t = lane; t < K; t += 32) out[t] = -1;
  int cnt = 0;
  for (int base = 0; base < N && cnt < K; base += 32) {
    int j = base + lane;
    bool w = false;
    if (j < N) {
      float dx = P[j * 3] - cx, dy = P[j * 3 + 1] - cy, dz = P[j * 3 + 2] - cz;
      w = (dx * dx + dy * dy + dz * dz) < r2;
    }
    unsigned m = (unsigned)__ballot(w);
    int off = cnt + __popc(m & ((1u << lane) - 1u));
    if (w && off < K) out[off] = j;
    cnt += __popc(m);
  }
}

// ---------------------------------------------------------------------------
// Grouping: X[b,s,k,:] = [feats[idx] (0 if pad), pts[idx]-centroid (-centroid if pad)]
// ---------------------------------------------------------------------------
__global__ void group_kernel(const float* __restrict__ pts,
                             const float* __restrict__ feats,
                             const float* __restrict__ cent,
                             const int* __restrict__ idx,
                             float* __restrict__ X,
                             int total, int S, int N, int K, int C) {
  int t = blockIdx.x * blockDim.x + threadIdx.x;
  if (t >= total) return;
  int bs = t / K;
  int b = bs / S;
  int i = idx[t];
  const float* cc = cent + (size_t)bs * 3;
  float* xr = X + (size_t)t * (C + 3);
  if (i >= 0) {
    const float* f = feats + ((size_t)b * N + i) * C;
    for (int c = 0; c < C; ++c) xr[c] = f[c];
    const float* p = pts + ((size_t)b * N + i) * 3;
    xr[C + 0] = p[0] - cc[0]; xr[C + 1] = p[1] - cc[1]; xr[C + 2] = p[2] - cc[2];
  } else {
    for (int c = 0; c < C; ++c) xr[c] = 0.0f;
    xr[C + 0] = -cc[0]; xr[C + 1] = -cc[1]; xr[C + 2] = -cc[2];
  }
}

// ---------------------------------------------------------------------------
// Fused 1x1-conv layer: Y = [relu]( (X[M,Ci] . W^T[Ci,Co]) * s + b ), f32 WMMA.
// Block = 128 threads = 4 wave32; tile = 128 rows x 16 cols; each wave owns
// two 16-row tiles so every B fragment is reused by two WMMAs (halves LDS
// weight traffic per MAC). K-chunks of 16 staged through LDS; next X chunk
// prefetched (global_prefetch_b8) to overlap L2 fill with WMMA issue.
// Wave32 fragment layouts per CDNA5 ISA 7.12.2:
//   A 16x4 : lanes 0-15 -> rows, K pair {0,1}; lanes 16-31 -> K pair {2,3}
//   B 4x16 : lane%16 -> col, lane half selects K pair (mirror of A)
//   C/D    : 8 VGPRs; lanes 0-15 rows 0-7, lanes 16-31 rows 8-15
// ---------------------------------------------------------------------------
__global__ __launch_bounds__(128) void mlp_layer_kernel(
    const float* __restrict__ X, const float* __restrict__ W,
    const float* __restrict__ sc, const float* __restrict__ bi,
    float* __restrict__ Y, int M, int Ci, int Co, int relu) {
  __shared__ float Xs[TROWS][18];   // padded: conflict-free column reads
  __shared__ float Ws[16][17];
  const int t = threadIdx.x;
  const int lane = t & 31, wave = t >> 5;
  const int l16 = lane & 15, half = lane >> 4;
  const int row0 = blockIdx.x * TROWS;
  const int col0 = blockIdx.y * 16;
  v8f acc0 = {0.f, 0.f, 0.f, 0.f, 0.f, 0.f, 0.f, 0.f};
  v8f acc1 = {0.f, 0.f, 0.f, 0.f, 0.f, 0.f, 0.f, 0.f};

  for (int k0 = 0; k0 < Ci; k0 += 16) {
    // Hint-prefetch part of the next X chunk into cache.
    if (k0 + 16 < Ci) {
      int pr = row0 + (t >> 4) * 8, pc = k0 + 16 + (t & 15);
      if (pr < M && pc < Ci) __builtin_prefetch(&X[(size_t)pr * Ci + pc], 0, 0);
    }
    // stage X tile: TROWS*16 elems
    for (int e = t; e < TROWS * 16; e += 128) {
      int r = e >> 4, c = e & 15;
      int gr = row0 + r, gc = k0 + c;
      Xs[r][c] = (gr < M && gc < Ci) ? X[(size_t)gr * Ci + gc] : 0.0f;
    }
    // stage W^T tile: 16x16
    for (int e = t; e < 16 * 16; e += 128) {
      int r = e >> 4, c = e & 15;
      int gk = k0 + r, gco = col0 + c;
      Ws[r][c] = (gk < Ci && gco < Co) ? W[(size_t)gco * Ci + gk] : 0.0f;
    }
    __syncthreads();
#pragma unroll
    for (int kk = 0; kk < 16; kk += 4) {
      int ko = kk + (half << 1);
      v2f bfr = {Ws[ko][l16], Ws[ko + 1][l16]};
      v2f a0 = {Xs[wave * 32 + l16][ko],      Xs[wave * 32 + l16][ko + 1]};
      v2f a1 = {Xs[wave * 32 + 16 + l16][ko], Xs[wave * 32 + 16 + l16][ko + 1]};
      acc0 = __builtin_amdgcn_wmma_f32_16x16x4_f32(
          false, a0, false, bfr, (short)0, acc0, false, false);
      acc1 = __builtin_amdgcn_wmma_f32_16x16x4_f32(
          false, a1, false, bfr, (short)0, acc1, false, false);
    }
    __syncthreads();
  }

  const int col = col0 + l16;
  float s = 1.0f, bb = 0.0f;
  if (sc != nullptr && col < Co) { s = sc[col]; bb = bi[col]; }
  const int rbase = row0 + wave * 32 + (half ? 8 : 0);
#pragma unroll
  for (int j = 0; j < 8; ++j) {
    int row = rbase + j;
    float v = acc0[j];
    if (sc != nullptr) v = v * s + bb;
    if (relu) v = fmaxf(v, 0.0f);
    if (row < M && col < Co) Y[(size_t)row * Co + col] = v;
  }
#pragma unroll
  for (int j = 0; j < 8; ++j) {
    int row = rbase + 16 + j;
    float v = acc1[j];
    if (sc != nullptr) v = v * s + bb;
    if (relu) v = fmaxf(v, 0.0f);
    if (row < M && col < Co) Y[(size_t)row * Co + col] = v;
  }
}

// ---------------------------------------------------------------------------
// Max over neighbors K, written into the concatenated feature buffer.
// ---------------------------------------------------------------------------
__global__ void maxpool_kernel(const float* __restrict__ Yin, float* __restrict__ Fout,
                               int total, int K, int C, int Ctot, int coff) {
  int t = blockIdx.x * blockDim.x + threadIdx.x;
  if (t >= total) return;
  int c = t % C, bs = t / C;
  const float* y = Yin + (size_t)bs * K * C + c;
  float m = y[0];
  for (int k = 1; k < K; ++k) m = fmaxf(m, y[(size_t)k * C]);
  Fout[(size_t)bs * Ctot + coff + c] = m;
}

// ---------------------------------------------------------------------------
// 3-NN on squared distances (reference interpolates with d^2, not d),
// stable tie-break (strict <), weights 1/(d2+1e-8) normalized.
// ---------------------------------------------------------------------------
__global__ void knn3_kernel(const float* __restrict__ p1, const float* __restrict__ p2,
                            int* __restrict__ ni, float* __restrict__ nw,
                            int total, int N1, int S2) {
  int t = blockIdx.x * blockDim.x + threadIdx.x;
  if (t >= total) return;
  int b = t / N1;
  const float* q = p1 + (size_t)t * 3;
  const float* Q = p2 + (size_t)b * S2 * 3;
  float qx = q[0], qy = q[1], qz = q[2];
  float d0 = 3.4e38f, d1 = 3.4e38f, d2 = 3.4e38f;
  int i0 = 0, i1 = 0, i2 = 0;
  for (int s = 0; s < S2; ++s) {
    float dx = Q[s * 3] - qx, dy = Q[s * 3 + 1] - qy, dz = Q[s * 3 + 2] - qz;
    float dd = dx * dx + dy * dy + dz * dz;
    if (dd < d0)      { d2 = d1; i2 = i1; d1 = d0; i1 = i0; d0 = dd; i0 = s; }
    else if (dd < d1) { d2 = d1; i2 = i1; d1 = dd; i1 = s; }
    else if (dd < d2) { d2 = dd; i2 = s; }
  }
  float r0 = 1.0f / (d0 + 1e-8f), r1 = 1.0f / (d1 + 1e-8f), r2v = 1.0f / (d2 + 1e-8f);
  float sm = r0 + r1 + r2v;
  nw[t * 3 + 0] = r0 / sm; nw[t * 3 + 1] = r1 / sm; nw[t * 3 + 2] = r2v / sm;
  ni[t * 3 + 0] = i0; ni[t * 3 + 1] = i1; ni[t * 3 + 2] = i2;
}

// ---------------------------------------------------------------------------
// FP input: X = [f1, sum_i w_i * f2[nn_i]]
// ---------------------------------------------------------------------------
__global__ void buildfp_kernel(const float* __restrict__ f1, const float* __restrict__ f2,
                               const int* __restrict__ ni, const float* __restrict__ nw,
                               float* __restrict__ X, int total, int N1, int S2,
                               int C1, int C2) {
  int t = blockIdx.x * blockDim.x + threadIdx.x;
  if (t >= total) return;
  int b = t / N1;
  float* xr = X + (size_t)t * (C1 + C2);
  const float* f = f1 + (size_t)t * C1;
  for (int c = 0; c < C1; ++c) xr[c] = f[c];
  const float* F2 = f2 + (size_t)b * S2 * C2;
  const float* A = F2 + (size_t)ni[t * 3 + 0] * C2;
  const float* Bp = F2 + (size_t)ni[t * 3 + 1] * C2;
  const float* Cp = F2 + (size_t)ni[t * 3 + 2] * C2;
  float wa = nw[t * 3 + 0], wb = nw[t * 3 + 1], wc = nw[t * 3 + 2];
  for (int c = 0; c < C2; ++c) xr[C1 + c] = wa * A[c] + wb * Bp[c] + wc * Cp[c];
}

// ---------------------------------------------------------------------------
// Host orchestration
// ---------------------------------------------------------------------------
struct LayerP { const float* W; const float* s; const float* b; int ci, co; };

extern "C" void kernel_launch(void* const* d_in, const int* in_sizes, int n_in,
                              void* d_out, int out_size, void* d_ws, size_t ws_size,
                              hipStream_t stream) {
  (void)in_sizes; (void)n_in; (void)out_size; (void)ws_size;
  const float* pts  = (const float*)d_in[0];
  const float* mask = (const float*)d_in[1];
  const float* nrm  = (const float*)d_in[2];

  // Params in setup_inputs() insertion order: sa -> fp -> out, each layer W,s,b.
  int cur = 3;
  auto nextL = [&](int co, int ci) {
    LayerP L;
    L.W = (const float*)d_in[cur]; L.s = (const float*)d_in[cur + 1];
    L.b = (const float*)d_in[cur + 2]; cur += 3; L.ci = ci; L.co = co;
    return L;
  };

  static const int   NP[5] = {4096, 1024, 256, 64, 16};
  static const int   CF[5] = {4, 96, 256, 512, 1024};
  static const float SA_R[4][2] = {{0.05f, 0.1f}, {0.1f, 0.2f}, {0.2f, 0.4f}, {0.4f, 0.8f}};
  static const int   SA_K[2] = {16, 32};
  static const int   SA_CH[4][2][3] = {
      {{16, 16, 32},   {32, 32, 64}},
      {{64, 64, 128},  {64, 96, 128}},
      {{128, 196, 256},{128, 196, 256}},
      {{256, 256, 512},{256, 384, 512}}};
  static const int FP_CIN[4] = {1536, 512, 352, 132};
  static const int FP_CH[4][3] = {{256, 256, 0}, {256, 256, 0}, {256, 128, 0}, {128, 128, 128}};
  static const int FP_NL[4] = {2, 2, 2, 3};

  LayerP saL[4][2][3];
  for (int l = 0; l < 4; ++l)
    for (int br = 0; br < 2; ++br) {
      int ci = CF[l] + 3;
      for (int li = 0; li < 3; ++li) { saL[l][br][li] = nextL(SA_CH[l][br][li], ci); ci = SA_CH[l][br][li]; }
    }
  LayerP fpL[4][3];
  for (int j = 0; j < 4; ++j) {
    int ci = FP_CIN[j];
    for (int li = 0; li < FP_NL[j]; ++li) { fpL[j][li] = nextL(FP_CH[j][li], ci); ci = FP_CH[j][li]; }
  }
  LayerP out0 = nextL(128, 128), out1 = nextL(128, 128);
  LayerP out2; out2.W = (const float*)d_in[cur++]; out2.s = nullptr; out2.b = nullptr;
  out2.ci = 128; out2.co = 3;

  // ---- workspace bump allocator ----
  char* wsp = (char*)d_ws;
  size_t off = 0;
  auto alloc = [&](size_t nfloats) -> float* {
    float* p = (float*)(wsp + off);
    off += ((nfloats * 4 + 255) & ~(size_t)255);
    return p;
  };
  float* f0 = alloc((size_t)B_SZ * N0 * 4);
  float* Pw[5]; float* F[5];
  Pw[0] = (float*)pts;   // never written
  F[0] = f0;
  for (int l = 1; l <= 4; ++l) {
    Pw[l] = alloc((size_t)B_SZ * NP[l] * 3);
    F[l]  = alloc((size_t)B_SZ * NP[l] * CF[l]);
  }
  int*   idxb  = (int*)alloc((size_t)B_SZ * 1024 * 32);
  int*   nnI   = (int*)alloc((size_t)B_SZ * N0 * 3);
  float* nnW   = alloc((size_t)B_SZ * N0 * 3);
  float* arenaA = alloc((size_t)8 * 1024 * 1024 + 131072);   // >= 262144*32
  float* arenaB = alloc((size_t)16 * 1024 * 1024 + 131072);  // >= 262144*64
  float* fpb[2];
  fpb[0] = alloc((size_t)B_SZ * N0 * 128);
  fpb[1] = alloc((size_t)B_SZ * N0 * 128);

  auto gemm = [&](const float* X, const LayerP& L, float* Y, int M, int relu) {
    dim3 g((unsigned)((M + TROWS - 1) / TROWS), (unsigned)((L.co + 15) / 16));
    mlp_layer_kernel<<<g, dim3(128), 0, stream>>>(X, L.W, L.s, L.b, Y, M, L.ci, L.co, relu);
  };

  // ---- encoder ----
  buildf0_kernel<<<(B_SZ * N0 + 255) / 256, 256, 0, stream>>>(mask, nrm, f0, B_SZ * N0);

  for (int l = 0; l < 4; ++l) {
    int Nl = NP[l], S = NP[l + 1];
    fps_kernel<<<B_SZ, FPS_T, 0, stream>>>(Pw[l], Pw[l + 1], Nl, S);
    int coff = 0;
    for (int br = 0; br < 2; ++br) {
      int K = SA_K[br];
      float r = SA_R[l][br];
      int tot = B_SZ * S;
      ballq_kernel<<<(tot + 7) / 8, 256, 0, stream>>>(Pw[l + 1], Pw[l], idxb, tot, S, Nl, r * r, K);
      int M = tot * K;
      group_kernel<<<(M + 255) / 256, 256, 0, stream>>>(Pw[l], F[l], Pw[l + 1], idxb, arenaA,
                                                        M, S, Nl, K, CF[l]);
      const float* src = arenaA;
      float* pp[2] = {arenaB, arenaA};
      for (int li = 0; li < 3; ++li) {
        float* dst = pp[li & 1];
        gemm(src, saL[l][br][li], dst, M, 1);
        src = dst;
      }
      int Cl = SA_CH[l][br][2];
      maxpool_kernel<<<(tot * Cl + 255) / 256, 256, 0, stream>>>(src, F[l + 1], tot * Cl,
                                                                 K, Cl, CF[l + 1], coff);
      coff += Cl;
    }
  }

  // ---- decoder (feature propagation) ----
  const float* feats = F[4];
  int Cfe = CF[4];
  int fpsel = 0;
  for (int j = 0; j < 4; ++j) {
    int lvl = 3 - j;
    int N1 = NP[lvl], S2 = NP[lvl + 1];
    int tot = B_SZ * N1;
    knn3_kernel<<<(tot + 255) / 256, 256, 0, stream>>>(Pw[lvl], Pw[lvl + 1], nnI, nnW, tot, N1, S2);
    buildfp_kernel<<<(tot + 255) / 256, 256, 0, stream>>>(F[lvl], feats, nnI, nnW, arenaA,
                                                          tot, N1, S2, CF[lvl], Cfe);
    const float* src = arenaA;
    float* pp[2] = {arenaB, arenaA};
    int nl = FP_NL[j];
    for (int li = 0; li < nl; ++li) {
      float* dst = (li == nl - 1) ? fpb[fpsel] : pp[li & 1];
      gemm(src, fpL[j][li], dst, tot, 1);
      src = dst;
    }
    feats = fpb[fpsel];
    fpsel ^= 1;
    Cfe = FP_CH[j][nl - 1];
  }

  // ---- head ----
  int M = B_SZ * N0;
  gemm(feats, out0, arenaA, M, 1);
  gemm(arenaA, out1, arenaB, M, 1);
  gemm(arenaB, out2, (float*)d_out, M, 0);   // final linear: no scale/bias/relu
}